// ICELUT_57123065037165
// MI455X (gfx1250) — compile-verified
//
#include <hip/hip_runtime.h>
#include <hip/hip_bf16.h>

typedef float v2f __attribute__((ext_vector_type(2)));
typedef float v8f __attribute__((ext_vector_type(8)));

#define HH 2048
#define WW 2048
#define HWSZ (HH * WW)          // 4,194,304
#define DIMC 33
#define D2 (DIMC * DIMC)        // 1089
#define D3 (DIMC * D2)          // 35937
#define NUML 20
#define NBLK_A 2048

// Padded GEMM shapes for the WMMA kernel (M=300->304, K=10->12, N=1089->1104)
#define MP 304
#define KP 12
#define NP 1104

// ---------------------------------------------------------------------------
// Phase A: gather-reduce mean of feature_msb[idx_m] + feature_lsb[idx_l]
// over 4M pixels, 20 channels. Deterministic two-level reduction.
// ---------------------------------------------------------------------------
__global__ __launch_bounds__(256) void k_feat_reduce(
    const int* __restrict__ msb, const int* __restrict__ lsb,
    const float* __restrict__ fm, const float* __restrict__ fl,
    float* __restrict__ partial)
{
    __shared__ float red[256 * NUML];
    float acc[NUML];
#pragma unroll
    for (int c = 0; c < NUML; ++c) acc[c] = 0.0f;

    const int w0 = 16 * 17 * 17, w1 = 16 * 17, w2 = 16;
    const int tid = blockIdx.x * blockDim.x + threadIdx.x;

    for (int g = 0; g < 2; ++g) {
        int p = tid * 8 + g * 4;
        int4 m0 = *(const int4*)(msb + p);
        int4 m1 = *(const int4*)(msb + HWSZ + p);
        int4 m2 = *(const int4*)(msb + 2 * HWSZ + p);
        int4 l0 = *(const int4*)(lsb + p);
        int4 l1 = *(const int4*)(lsb + HWSZ + p);
        int4 l2 = *(const int4*)(lsb + 2 * HWSZ + p);

        int im[4] = { m0.x * w0 + m1.x * w1 + m2.x * w2,
                      m0.y * w0 + m1.y * w1 + m2.y * w2,
                      m0.z * w0 + m1.z * w1 + m2.z * w2,
                      m0.w * w0 + m1.w * w1 + m2.w * w2 };
        int il[4] = { l0.x * w0 + l1.x * w1 + l2.x * w2,
                      l0.y * w0 + l1.y * w1 + l2.y * w2,
                      l0.z * w0 + l1.z * w1 + l2.z * w2,
                      l0.w * w0 + l1.w * w1 + l2.w * w2 };

#pragma unroll
        for (int q = 0; q < 4; ++q) {
            const float4* rm = (const float4*)(fm + (size_t)im[q] * NUML);
            const float4* rl = (const float4*)(fl + (size_t)il[q] * NUML);
#pragma unroll
            for (int v = 0; v < 5; ++v) {
                float4 a = rm[v];
                float4 b = rl[v];
                acc[v * 4 + 0] += a.x + b.x;
                acc[v * 4 + 1] += a.y + b.y;
                acc[v * 4 + 2] += a.z + b.z;
                acc[v * 4 + 3] += a.w + b.w;
            }
        }
    }

    int t = threadIdx.x;
#pragma unroll
    for (int c = 0; c < NUML; ++c) red[t * NUML + c] = acc[c];
    __syncthreads();
    if (t < NUML) {
        float s = 0.0f;
        for (int i = 0; i < 256; ++i) s += red[i * NUML + t];
        partial[blockIdx.x * NUML + t] = s;
    }
}

// ---------------------------------------------------------------------------
// Phase B1: final mean, quantize, build pair indices, gather lut_cat rows,
// produce weights[20]. Single wave, fixed-order sums -> deterministic.
// ---------------------------------------------------------------------------
__global__ __launch_bounds__(32) void k_weights(
    const float* __restrict__ partial, const float* __restrict__ lut_cat,
    float* __restrict__ wout)
{
    __shared__ int midi[NUML];
    int t = threadIdx.x;
    if (t < NUML) {
        float s = 0.0f;
        for (int i = 0; i < NBLK_A; ++i) s += partial[i * NUML + t];
        float mean = s * (1.0f / (float)HWSZ);
        float mid = rintf(mean * 4.0f) * 0.25f;        // round-half-even like jnp.round
        mid = fminf(fmaxf(mid, -32.0f), 31.75f);
        midi[t] = (int)(mid * 4.0f) + 128;             // exact integer, trunc == value
    }
    __syncthreads();
    if (t < NUML) {
        float acc = 0.0f;
        for (int k = 0; k < 10; ++k) {
            int index = midi[2 * k] * 256 + midi[2 * k + 1];
            acc += lut_cat[((size_t)k * 65536 + index) * NUML + t];
        }
        wout[t] = acc;
    }
}

// ---------------------------------------------------------------------------
// Phase B2a: zero-pad luts (300x10 -> 304x12) and w_layers (10x1089 ->
// 12x1104) into workspace so the WMMA wave runs branch-free.
// ---------------------------------------------------------------------------
__global__ __launch_bounds__(256) void k_pad(
    const float* __restrict__ A, const float* __restrict__ B,
    float* __restrict__ Apad, float* __restrict__ Bpad)
{
    int i = blockIdx.x * blockDim.x + threadIdx.x;
    if (i < MP * KP) {
        int r = i / KP, k = i % KP;
        Apad[i] = (r < 300 && k < 10) ? A[r * 10 + k] : 0.0f;
    }
    if (i < KP * NP) {
        int k = i / NP, col = i % NP;
        Bpad[i] = (k < 10 && col < D2) ? B[k * D2 + col] : 0.0f;
    }
}

// ---------------------------------------------------------------------------
// Phase B2b (WMMA): Cpad(304x1104) = Apad(304x12) @ Bpad(12x1104), f32 via
// V_WMMA_F32_16X16X4_F32. One wave32 per 16x16 tile, 3 chained k-steps,
// no branches, no exec-mask churn.
// A 16x4 layout: lanes 0-15 hold K=k0,k0+1 for row M=lane&15; lanes 16-31
// hold K=k0+2,k0+3. B mirrored (col = lane&15). C/D: vgpr r -> M=r+8*half.
// ---------------------------------------------------------------------------
__global__ __launch_bounds__(32) void k_wmma_m(
    const float* __restrict__ Ap, const float* __restrict__ Bp,
    float* __restrict__ Cp)
{
    int tileN = blockIdx.x % (NP / 16);
    int tileM = blockIdx.x / (NP / 16);
    int m0 = tileM * 16, n0 = tileN * 16;
    int lane = threadIdx.x;
    int half = lane >> 4;
    int l = lane & 15;
    int row = m0 + l;
    int col = n0 + l;

    v8f c = {};
#pragma unroll
    for (int k0 = 0; k0 < KP; k0 += 4) {
        int ka = k0 + half * 2;
        v2f a = *(const v2f*)(Ap + row * KP + ka);     // contiguous, 8B aligned
        v2f b;
        b.x = Bp[ka * NP + col];
        b.y = Bp[(ka + 1) * NP + col];
        c = __builtin_amdgcn_wmma_f32_16x16x4_f32(
            /*neg_a=*/false, a, /*neg_b=*/false, b,
            /*c_mod=*/(short)0, c, /*reuse_a=*/false, /*reuse_b=*/false);
    }

#pragma unroll
    for (int r = 0; r < 8; ++r)
        Cp[(m0 + half * 8 + r) * NP + col] = c[r];
}

// ---------------------------------------------------------------------------
// Phase B3: contract the 20-LUT axis with weights (push weights through the
// linear reconstruction): mw[s][c][j] = sum_n w[n] * Cpad[s*60+n*3+c][j]
// ---------------------------------------------------------------------------
__global__ __launch_bounds__(256) void k_mw(
    const float* __restrict__ m, const float* __restrict__ w,
    float* __restrict__ mw)
{
    int i = blockIdx.x * blockDim.x + threadIdx.x;
    if (i >= 5 * 3 * D2) return;
    int j = i % D2;
    int sc = i / D2;
    int c = sc % 3, s = sc / 3;
    float acc = 0.0f;
#pragma unroll
    for (int n = 0; n < NUML; ++n)
        acc += w[n] * m[(s * 60 + n * 3 + c) * NP + j];
    mw[i] = acc;
}

// ---------------------------------------------------------------------------
// Phase B4: d3lut[c][a][b][e] = sum_s s_layers[z][s] * mw[s][c][y*33+x]
// with the per-channel cube_to_lut permutation folded into (z,y,x):
//   c=0: (z,y,x)=(e,a,b)   c=1: (z,y,x)=(a,e,b)   c=2: (z,y,x)=(a,b,e)
// ---------------------------------------------------------------------------
__global__ __launch_bounds__(256) void k_d3lut(
    const float* __restrict__ sl,  // (33,5)
    const float* __restrict__ mw,  // (5,3,1089)
    float* __restrict__ out)       // (3,33,33,33)
{
    int f = blockIdx.x * blockDim.x + threadIdx.x;
    if (f >= 3 * D3) return;
    int c = f / D3;
    int r = f % D3;
    int a = r / D2;
    int r2 = r % D2;
    int b = r2 / DIMC;
    int e = r2 % DIMC;
    int z, y, x;
    if (c == 0)      { z = e; y = a; x = b; }
    else if (c == 1) { z = a; y = e; x = b; }
    else             { z = a; y = b; x = e; }
    float acc = 0.0f;
#pragma unroll
    for (int s = 0; s < 5; ++s)
        acc += sl[z * 5 + s] * mw[(s * 3 + c) * D2 + y * DIMC + x];
    out[f] = acc;
}

// ---------------------------------------------------------------------------
// Phase C: trilinear interpolation + residual add. 4 pixels/thread with
// float4 loads/stores; LUT (431 KB) is L2-resident.
// ---------------------------------------------------------------------------
__global__ __launch_bounds__(256) void k_trilinear(
    const float* __restrict__ img, const float* __restrict__ lut,
    float* __restrict__ out)
{
    int p = (blockIdx.x * 256 + threadIdx.x) * 4;
    float4 R = *(const float4*)(img + p);
    float4 G = *(const float4*)(img + HWSZ + p);
    float4 B = *(const float4*)(img + 2 * HWSZ + p);

    float rr[4] = { R.x, R.y, R.z, R.w };
    float gg[4] = { G.x, G.y, G.z, G.w };
    float bb[4] = { B.x, B.y, B.z, B.w };
    float orr[4], ogg[4], obb[4];

    const float invbin = (float)(DIMC - 1) / 1.000001f;

#pragma unroll
    for (int q = 0; q < 4; ++q) {
        float fr = rr[q] * invbin;
        float fg = gg[q] * invbin;
        float fb = bb[q] * invbin;
        int ir = (int)floorf(fr); ir = ir < 0 ? 0 : (ir > DIMC - 2 ? DIMC - 2 : ir);
        int ig = (int)floorf(fg); ig = ig < 0 ? 0 : (ig > DIMC - 2 ? DIMC - 2 : ig);
        int ib = (int)floorf(fb); ib = ib < 0 ? 0 : (ib > DIMC - 2 ? DIMC - 2 : ib);
        float rd = fr - (float)ir;
        float gd = fg - (float)ig;
        float bd = fb - (float)ib;

        int base = ib * D2 + ig * DIMC + ir;
        float res0 = 0.0f, res1 = 0.0f, res2 = 0.0f;
#pragma unroll
        for (int i = 0; i < 2; ++i) {
            float wb = i ? bd : 1.0f - bd;
#pragma unroll
            for (int j = 0; j < 2; ++j) {
                float wg = j ? gd : 1.0f - gd;
                float wlo = wb * wg * (1.0f - rd);
                float whi = wb * wg * rd;
                int off = base + i * D2 + j * DIMC;
                res0 += wlo * lut[off]          + whi * lut[off + 1];
                res1 += wlo * lut[D3 + off]     + whi * lut[D3 + off + 1];
                res2 += wlo * lut[2 * D3 + off] + whi * lut[2 * D3 + off + 1];
            }
        }
        orr[q] = res0 + rr[q];
        ogg[q] = res1 + gg[q];
        obb[q] = res2 + bb[q];
    }

    *(float4*)(out + p)            = make_float4(orr[0], orr[1], orr[2], orr[3]);
    *(float4*)(out + HWSZ + p)     = make_float4(ogg[0], ogg[1], ogg[2], ogg[3]);
    *(float4*)(out + 2 * HWSZ + p) = make_float4(obb[0], obb[1], obb[2], obb[3]);
}

// ---------------------------------------------------------------------------
extern "C" void kernel_launch(void* const* d_in, const int* in_sizes, int n_in,
                              void* d_out, int out_size, void* d_ws, size_t ws_size,
                              hipStream_t stream)
{
    const int*   img_msb  = (const int*)d_in[0];
    const int*   img_lsb  = (const int*)d_in[1];
    const float* img_org  = (const float*)d_in[2];
    const float* feat_m   = (const float*)d_in[3];
    const float* feat_l   = (const float*)d_in[4];
    const float* lut_cat  = (const float*)d_in[5];
    const float* s_layers = (const float*)d_in[6];
    const float* w_layers = (const float*)d_in[7];
    const float* luts     = (const float*)d_in[8];

    float* out = (float*)d_out;
    float* ws  = (float*)d_ws;

    float* ws_sums = ws;                               // 2048*20      = 40960
    float* ws_w    = ws_sums + NBLK_A * NUML;          // 20
    float* ws_Ap   = ws_w + NUML;                      // 304*12       = 3648
    float* ws_Bp   = ws_Ap + MP * KP;                  // 12*1104      = 13248
    float* ws_Cp   = ws_Bp + KP * NP;                  // 304*1104     = 335616
    float* ws_mw   = ws_Cp + MP * NP;                  // 5*3*1089     = 16335

    float* d3lut_out = out + (size_t)3 * HWSZ;         // second tuple output

    k_feat_reduce<<<NBLK_A, 256, 0, stream>>>(img_msb, img_lsb, feat_m, feat_l, ws_sums);
    k_weights    <<<1, 32, 0, stream>>>(ws_sums, lut_cat, ws_w);
    k_pad        <<<(KP * NP + 255) / 256, 256, 0, stream>>>(luts, w_layers, ws_Ap, ws_Bp);
    k_wmma_m     <<<(MP / 16) * (NP / 16), 32, 0, stream>>>(ws_Ap, ws_Bp, ws_Cp);
    k_mw         <<<(5 * 3 * D2 + 255) / 256, 256, 0, stream>>>(ws_Cp, ws_w, ws_mw);
    k_d3lut      <<<(3 * D3 + 255) / 256, 256, 0, stream>>>(s_layers, ws_mw, d3lut_out);
    k_trilinear  <<<HWSZ / 4 / 256, 256, 0, stream>>>(img_org, d3lut_out, out);
}